// onering_conv_layer_45500883534540
// MI455X (gfx1250) — compile-verified
//
#include <hip/hip_runtime.h>

typedef __attribute__((ext_vector_type(2))) float v2f;
typedef __attribute__((ext_vector_type(8))) float v8f;

#define IN_F   64
#define OUT_F  64
#define KNB    7
#define KDIM   448                 // 7 * 64
#define APAD   452                 // padded LDS row stride in floats (448+4: bank-conflict free)
#define TILES_PER_BLOCK 8
#define NODES_PER_BLOCK (16 * TILES_PER_BLOCK)   // 128 nodes per workgroup
#define THREADS 128                               // 4 wave32

// Burst of N async 16B global->LDS transfers from one base address pair.
// INST_OFFSET is added to BOTH the global address and the LDS address (ISA 10.x),
// so a single VGPR address setup serves the whole contiguous run.
template <int N> struct Burst {
    static __device__ __forceinline__ void run(unsigned lds_off, const char* gaddr) {
        Burst<N - 1>::run(lds_off, gaddr);
        asm volatile("global_load_async_to_lds_b128 %0, %1, off offset:%2"
                     :: "v"(lds_off), "v"(gaddr), "i"((N - 1) * 16)
                     : "memory");
    }
};
template <> struct Burst<0> {
    static __device__ __forceinline__ void run(unsigned, const char*) {}
};

__device__ __forceinline__ void wait_asynccnt0() {
    asm volatile("s_wait_asynccnt 0x0" ::: "memory");
}

__device__ __forceinline__ v8f wmma_f32(v2f a, v2f b, v8f c) {
    return __builtin_amdgcn_wmma_f32_16x16x4_f32(
        /*neg_a=*/false, a, /*neg_b=*/false, b,
        /*c_mod=*/(short)0, c, /*reuse_a=*/false, /*reuse_b=*/false);
}

__global__ __launch_bounds__(THREADS, 1)
void onering_conv_wmma_kernel(const float* __restrict__ x,
                              const long long* __restrict__ nbr,
                              const float* __restrict__ W,
                              const float* __restrict__ bias,
                              float* __restrict__ out,
                              int nNodes)
{
    // 64 * 452 * 4 = 115712 B  +  16 * 452 * 4 = 28928 B  -> 144640 B (2 blocks / 320 KB WGP)
    __shared__ float sW[OUT_F * APAD];
    __shared__ float sA[16 * APAD];

    const int tid  = threadIdx.x;
    const int lane = tid & 31;
    const int wave = tid >> 5;                 // 0..3, each owns 16 output columns

    // ---- stage W [64][448] into LDS once per block ----
    // thread owns half of row n: 56 async 16B transfers from one base address
    {
        const int n = tid >> 1;
        const int h = tid & 1;                 // which half of the row
        const char* src = (const char*)(W + n * KDIM + h * (KDIM / 2));
        const unsigned dst = (unsigned)(size_t)&sW[n * APAD + h * (KDIM / 2)];
        Burst<(KDIM / 2) / 4>::run(dst, src);  // 56 x b128
    }

    // per-lane fragment coordinates (WMMA f32 16x16x4 layout):
    //   half-wave h = lane>>4; element row/col = lane&15; K pair base = h*2
    const int ncol = wave * 16 + (lane & 15);  // output feature column
    const int koff = (lane >> 4) * 2;          // K offset of this lane's v2f pair
    const float bval = bias[ncol];

    const long long base_node = (long long)blockIdx.x * NODES_PER_BLOCK;
    const bool full_block = (base_node + NODES_PER_BLOCK) <= (long long)nNodes;  // uniform
    const long long max_nidx = (long long)nNodes * KNB - 1;

    // gather role: one thread per (m, j) row of the A tile (112 of 128 threads)
    const int g_m = tid / KNB;                 // 0..18 (only tid<112 used)
    const int g_j = tid - g_m * KNB;           // 0..6
    const unsigned g_dst = (unsigned)(size_t)&sA[g_m * APAD + g_j * IN_F];

    for (int t = 0; t < TILES_PER_BLOCK; ++t) {
        const long long tile_node = base_node + t * 16;

        // protects sA reuse across tiles (GEMM reads of tile t-1 are done)
        __syncthreads();

        // ---- gather: 16 nodes x 7 neighbors x 64 feats -> sA[16][448(+pad)] ----
        // one thread per row: 1 index load + 16 async b128 issues from one base
        if (tid < 16 * KNB) {
            long long nidx = (tile_node + g_m) * KNB + g_j;
            nidx = (nidx < max_nidx) ? nidx : max_nidx;     // branch-free tail clamp
            const long long src = nbr[nidx];                // int64 neighbor id in [0, N)
            const char* gsrc = (const char*)(x + src * IN_F);
            Burst<IN_F / 4>::run(g_dst, gsrc);              // 16 x b128 = 256 B row
        }
        // own async LDS writes (W stage at t==0 + this tile's gather) must land
        wait_asynccnt0();
        __syncthreads();

        // ---- GEMM: D[16x16] = A[16x448] * B[448x16] + bias ----
        // 4 independent accumulator chains so WMMAs pipeline instead of
        // serializing through a single C->D dependency.
        v8f c0 = { bval, bval, bval, bval, bval, bval, bval, bval };
        v8f c1 = {}, c2 = {}, c3 = {};
        const float* pA = &sA[(lane & 15) * APAD + koff];
        const float* pB = &sW[ncol * APAD + koff];

        #pragma unroll 2
        for (int k0 = 0; k0 < KDIM; k0 += 16) {
            v2f a0 = *(const v2f*)(pA + k0);
            v2f b0 = *(const v2f*)(pB + k0);
            v2f a1 = *(const v2f*)(pA + k0 + 4);
            v2f b1 = *(const v2f*)(pB + k0 + 4);
            v2f a2 = *(const v2f*)(pA + k0 + 8);
            v2f b2 = *(const v2f*)(pB + k0 + 8);
            v2f a3 = *(const v2f*)(pA + k0 + 12);
            v2f b3 = *(const v2f*)(pB + k0 + 12);
            c0 = wmma_f32(a0, b0, c0);
            c1 = wmma_f32(a1, b1, c1);
            c2 = wmma_f32(a2, b2, c2);
            c3 = wmma_f32(a3, b3, c3);
        }
        const v8f c = (c0 + c1) + (c2 + c3);

        // ---- store: lane holds D[r + (lane>>4)*8][ncol], r = 0..7 ----
        const long long mbase = tile_node + (long long)(lane >> 4) * 8;
        if (full_block) {
            #pragma unroll
            for (int r = 0; r < 8; ++r) {
                out[(mbase + r) * OUT_F + ncol] = c[r];
            }
        } else {
            #pragma unroll
            for (int r = 0; r < 8; ++r) {
                const long long node = mbase + r;
                if (node < nNodes) out[node * OUT_F + ncol] = c[r];
            }
        }
    }
}

extern "C" void kernel_launch(void* const* d_in, const int* in_sizes, int n_in,
                              void* d_out, int out_size, void* d_ws, size_t ws_size,
                              hipStream_t stream) {
    const float*     x    = (const float*)d_in[0];
    const long long* nbr  = (const long long*)d_in[1];
    const float*     W    = (const float*)d_in[2];
    const float*     bias = (const float*)d_in[3];
    float*           out  = (float*)d_out;

    const int nNodes = in_sizes[0] / IN_F;     // 163842
    const int grid   = (nNodes + NODES_PER_BLOCK - 1) / NODES_PER_BLOCK;

    onering_conv_wmma_kernel<<<grid, THREADS, 0, stream>>>(x, nbr, W, bias, out, nNodes);
}